// UserSocial_70892730188380
// MI455X (gfx1250) — compile-verified
//
#include <hip/hip_runtime.h>
#include <hip/hip_bf16.h>

#define EMB_DIM 128

typedef __attribute__((ext_vector_type(2))) float v2f;
typedef __attribute__((ext_vector_type(8))) float v8f;

// ---------------------------------------------------------------------------
// Native CDNA5 no-return f32 global atomic add (device scope). Inline asm so
// we deterministically get global_atomic_add_f32 (no CAS-loop fallback).
// ---------------------------------------------------------------------------
__device__ __forceinline__ void gatomic_add_f32(float* p, float v) {
  asm volatile("global_atomic_add_f32 %0, %1, off scope:SCOPE_DEV"
               :: "v"(p), "v"(v) : "memory");
}

__device__ __forceinline__ void gatomic_add_f32x4(float* p, float4 v) {
  asm volatile(
      "global_atomic_add_f32 %0, %1, off scope:SCOPE_DEV\n\t"
      "global_atomic_add_f32 %0, %2, off offset:4 scope:SCOPE_DEV\n\t"
      "global_atomic_add_f32 %0, %3, off offset:8 scope:SCOPE_DEV\n\t"
      "global_atomic_add_f32 %0, %4, off offset:12 scope:SCOPE_DEV"
      :: "v"(p), "v"(v.x), "v"(v.y), "v"(v.z), "v"(v.w) : "memory");
}

// ---------------------------------------------------------------------------
// Zero the workspace (deg + g1 + g2 accumulators).
// ---------------------------------------------------------------------------
__global__ void zero_f32(float* __restrict__ p, long long n) {
  long long t = (long long)blockIdx.x * blockDim.x + threadIdx.x;
  if (t < n) p[t] = 0.0f;
}

// ---------------------------------------------------------------------------
// Degree: one thread per edge. deg[] is 400KB -> lives in L2 (192MB).
// ---------------------------------------------------------------------------
__global__ void degree_kernel(const int* __restrict__ dst,
                              float* __restrict__ deg, int n_edges) {
  int e = blockIdx.x * blockDim.x + threadIdx.x;
  if (e < n_edges) gatomic_add_f32(deg + dst[e], 1.0f);
}

// ---------------------------------------------------------------------------
// Scatter conv: one wave32 per edge; each lane moves float4 (512B/row/wave).
// agg buffer (51.2MB) fits in MI455X's 192MB L2, so the atomic scatter is
// L2-resident instead of HBM round-trips.
// ---------------------------------------------------------------------------
__global__ void __launch_bounds__(256)
scatter_kernel(const int* __restrict__ src, const int* __restrict__ dst,
               const float* __restrict__ h, float* __restrict__ agg,
               int n_edges) {
  int t = blockIdx.x * blockDim.x + threadIdx.x;
  int e = t >> 5;
  int lane = t & 31;
  if (e >= n_edges) return;
  int s = src[e];
  int d = dst[e];
  const float4 v = *(const float4*)(h + (size_t)s * EMB_DIM + lane * 4);
  gatomic_add_f32x4(agg + (size_t)d * EMB_DIM + lane * 4, v);
}

// ---------------------------------------------------------------------------
// Mean normalize: g[u] /= max(deg[u], 1). One wave per user row.
// ---------------------------------------------------------------------------
__global__ void norm_kernel(float* __restrict__ g, const float* __restrict__ deg,
                            int num_users) {
  int t = blockIdx.x * blockDim.x + threadIdx.x;
  int u = t >> 5;
  int lane = t & 31;
  if (u >= num_users) return;
  float inv = 1.0f / fmaxf(deg[u], 1.0f);
  float4* p = (float4*)(g + (size_t)u * EMB_DIM + lane * 4);
  float4 v = *p;
  v.x *= inv; v.y *= inv; v.z *= inv; v.w *= inv;
  *p = v;
}

// ---------------------------------------------------------------------------
// Batched gather: latest_user = (emb + g1 + g2)[users], latest_item =
// item_emb[items]. One wave per batch row, float4 lanes, coalesced stores.
// ---------------------------------------------------------------------------
__global__ void __launch_bounds__(256)
gather_kernel(const int* __restrict__ users, const int* __restrict__ items,
              const float* __restrict__ uemb, const float* __restrict__ g1,
              const float* __restrict__ g2, const float* __restrict__ iemb,
              float* __restrict__ out_user, float* __restrict__ out_item,
              int batch) {
  int t = blockIdx.x * blockDim.x + threadIdx.x;
  int b = t >> 5;
  int lane = t & 31;
  if (b >= batch) return;
  int u = users[b];
  int it = items[b];
  size_t off = (size_t)lane * 4;
  const float4 a = *(const float4*)(uemb + (size_t)u * EMB_DIM + off);
  const float4 c = *(const float4*)(g1 + (size_t)u * EMB_DIM + off);
  const float4 d = *(const float4*)(g2 + (size_t)u * EMB_DIM + off);
  float4 fu;
  fu.x = a.x + c.x + d.x;
  fu.y = a.y + c.y + d.y;
  fu.z = a.z + c.z + d.z;
  fu.w = a.w + c.w + d.w;
  *(float4*)(out_user + (size_t)b * EMB_DIM + off) = fu;
  *(float4*)(out_item + (size_t)b * EMB_DIM + off) =
      *(const float4*)(iemb + (size_t)it * EMB_DIM + off);
}

// ---------------------------------------------------------------------------
// predict[b] = sigmoid(dot(latest_user[b], latest_item[b])) via WMMA.
// One wave32 handles 16 batch rows: accumulate D = U_tile x V_tile^T with
// 32 x V_WMMA_F32_16X16X4_F32 (K=4 f32, exact math), then read diag(D).
//
// A-operand layout (16x4 f32): lanes 0-15 carry M=lane {K=k0,k0+1},
// lanes 16-31 carry M=lane-16 {K=k0+2,k0+3} -> per lane: row = lane&15,
// k offset = k0 + 2*(lane>>4). B (4x16, = V^T chunk) mirrors with N=lane&15,
// so both operands load identically per lane.
// C/D layout: VGPR r holds M=r (lanes 0-15, N=lane) and M=r+8 (lanes 16-31,
// N=lane-16) -> diag element i sits at vector idx (lane&7), lane i<8?i:i+16.
// ---------------------------------------------------------------------------
__global__ void __launch_bounds__(256)
dot_sigmoid_wmma(const float* __restrict__ U, const float* __restrict__ V,
                 float* __restrict__ pred, int batch) {
  int gid = blockIdx.x * blockDim.x + threadIdx.x;
  int wave = gid >> 5;
  int lane = threadIdx.x & 31;
  if (wave * 16 >= batch) return;  // wave-uniform exit: EXEC all-1s for WMMA

  int row = lane & 15;
  int half = lane >> 4;
  const float* urow = U + ((size_t)wave * 16 + row) * EMB_DIM + 2 * half;
  const float* vrow = V + ((size_t)wave * 16 + row) * EMB_DIM + 2 * half;

  v8f acc = {};
#pragma unroll
  for (int k0 = 0; k0 < EMB_DIM; k0 += 4) {
    v2f a = *(const v2f*)(urow + k0);  // 8B-aligned float2 loads
    v2f b = *(const v2f*)(vrow + k0);
    acc = __builtin_amdgcn_wmma_f32_16x16x4_f32(
        /*neg_a=*/false, a, /*neg_b=*/false, b,
        /*c_mod=*/(short)0, acc, /*reuse_a=*/false, /*reuse_b=*/false);
  }

  // Diagonal extraction + sigmoid (lanes 0-7 hold diag 0-7, lanes 24-31 hold
  // diag 8-15; vector element index is lane&7 in both cases).
  if (lane < 8 || lane >= 24) {
    int i = (lane < 8) ? lane : (lane - 16);
    float d = acc[lane & 7];
    pred[(size_t)wave * 16 + i] = 1.0f / (1.0f + expf(-d));
  }
}

// ---------------------------------------------------------------------------
// Host-side launch. Inputs: users, items, edge_src, edge_dst, user_emb,
// item_emb. Output: [predict(B) | latest_user(B*128) | latest_item(B*128)].
// ---------------------------------------------------------------------------
extern "C" void kernel_launch(void* const* d_in, const int* in_sizes, int n_in,
                              void* d_out, int out_size, void* d_ws,
                              size_t ws_size, hipStream_t stream) {
  const int* users  = (const int*)d_in[0];
  const int* items  = (const int*)d_in[1];
  const int* esrc   = (const int*)d_in[2];
  const int* edst   = (const int*)d_in[3];
  const float* uemb = (const float*)d_in[4];
  const float* iemb = (const float*)d_in[5];

  const int batch     = in_sizes[0];
  const int n_edges   = in_sizes[2];
  const int num_users = in_sizes[4] / EMB_DIM;

  // Workspace layout: deg | g1 | g2
  float* deg = (float*)d_ws;
  float* g1  = deg + num_users;
  float* g2  = g1 + (size_t)num_users * EMB_DIM;

  // Output layout: predict | latest_user | latest_item
  float* pred     = (float*)d_out;
  float* out_user = pred + batch;
  float* out_item = out_user + (size_t)batch * EMB_DIM;

  const int B = 256;

  // 1) Zero deg + g1 + g2.
  long long total = (long long)num_users * (1 + 2 * EMB_DIM);
  zero_f32<<<(int)((total + B - 1) / B), B, 0, stream>>>((float*)d_ws, total);

  // 2) Degrees (shared by both conv layers).
  degree_kernel<<<(n_edges + B - 1) / B, B, 0, stream>>>(edst, deg, n_edges);

  // 3) Conv layer 1: scatter user_emb -> g1, then mean-normalize.
  long long sc_threads = (long long)n_edges * 32;
  scatter_kernel<<<(int)((sc_threads + B - 1) / B), B, 0, stream>>>(
      esrc, edst, uemb, g1, n_edges);
  long long nm_threads = (long long)num_users * 32;
  norm_kernel<<<(int)((nm_threads + B - 1) / B), B, 0, stream>>>(g1, deg,
                                                                 num_users);

  // 4) Conv layer 2: scatter g1 -> g2, then mean-normalize.
  scatter_kernel<<<(int)((sc_threads + B - 1) / B), B, 0, stream>>>(
      esrc, edst, g1, g2, n_edges);
  norm_kernel<<<(int)((nm_threads + B - 1) / B), B, 0, stream>>>(g2, deg,
                                                                 num_users);

  // 5) Batched gathers -> latest_user / latest_item.
  long long ga_threads = (long long)batch * 32;
  gather_kernel<<<(int)((ga_threads + B - 1) / B), B, 0, stream>>>(
      users, items, uemb, g1, g2, iemb, out_user, out_item, batch);

  // 6) WMMA batched dot + sigmoid -> predict.
  long long dot_threads = (long long)((batch + 15) / 16) * 32;
  dot_sigmoid_wmma<<<(int)((dot_threads + B - 1) / B), B, 0, stream>>>(
      out_user, out_item, pred, batch);
}